// GCNencoder_41480794145130
// MI455X (gfx1250) — compile-verified
//
#include <hip/hip_runtime.h>
#include <stdint.h>

typedef __attribute__((ext_vector_type(2))) float v2f;
typedef __attribute__((ext_vector_type(8))) float v8f;

#define IN_F 128
#define HID  64

// ---------------- degree / norm ----------------

__global__ void k_init_deg(float* __restrict__ deg, int N) {
    int i = blockIdx.x * blockDim.x + threadIdx.x;
    if (i < N) deg[i] = 1.0f;                      // self-loop contribution
}

__global__ void k_count_deg(const long long* __restrict__ dst,
                            float* __restrict__ deg, int E) {
    int e = blockIdx.x * blockDim.x + threadIdx.x;
    if (e < E) atomicAdd(&deg[(int)dst[e]], 1.0f);
}

__global__ void k_rsqrt(float* __restrict__ deg, int N) {
    int i = blockIdx.x * blockDim.x + threadIdx.x;
    if (i < N) deg[i] = rsqrtf(deg[i]);            // deg >= 1 always
}

// ---------------- WMMA GEMM: C[M,64] = A[M,K] x B[K,64] ----------------
// SPLITB: B column j < 32 comes from B0[k*32+j], else B1[k*32+j-32].
// One block = 128 threads = 4 waves; each wave owns one 16-col slice.
// Each block computes a 16x64 tile of C, staging A-tile and B in LDS.

template <int K, bool SPLITB>
__global__ __launch_bounds__(128) void k_gemm_wmma(
    const float* __restrict__ A, const float* __restrict__ B0,
    const float* __restrict__ B1, float* __restrict__ C) {
    __shared__ float sA[16 * K];
    __shared__ float sB[K * 64];
    const int t = threadIdx.x;
    const int mtile = blockIdx.x;

    // Stage A tile (16 x K), coalesced float4
    {
        const float4* a4 = (const float4*)(A + (size_t)mtile * 16 * K);
        float4* s4 = (float4*)sA;
        const int n4 = 16 * K / 4;
        for (int idx = t; idx < n4; idx += 128) s4[idx] = a4[idx];
    }
    // Stage B panel (K x 64), coalesced float4
    {
        float4* s4 = (float4*)sB;
        const int n4 = K * 64 / 4;
        for (int idx = t; idx < n4; idx += 128) {
            if (SPLITB) {
                int k  = idx >> 4;   // 16 float4 per 64-wide row
                int c4 = idx & 15;
                const float* p = (c4 < 8) ? (B0 + k * 32 + c4 * 4)
                                          : (B1 + k * 32 + (c4 - 8) * 4);
                s4[idx] = *(const float4*)p;
            } else {
                s4[idx] = ((const float4*)B0)[idx];
            }
        }
    }
    __syncthreads();

    const int lane = t & 31;
    const int wave = t >> 5;
    const int half = lane >> 4;   // 0: K pair {k0,k0+1}, 1: {k0+2,k0+3}
    const int l    = lane & 15;
    const int col0 = wave * 16;

    v8f c = {};
    const float* aRow  = sA + l * K + 2 * half;          // A[m=l][k0 + 2*half ..]
    const float* bBase = sB + 2 * half * 64 + col0 + l;  // B[k0 + 2*half][n]
#pragma unroll
    for (int k0 = 0; k0 < K; k0 += 4) {
        v2f a = *(const v2f*)(aRow + k0);
        v2f b;
        b.x = bBase[k0 * 64];
        b.y = bBase[k0 * 64 + 64];
        // D = A*B + C, fp32 in / fp32 out (precision-exact for this fp32 model)
        c = __builtin_amdgcn_wmma_f32_16x16x4_f32(false, a, false, b,
                                                  (short)0, c, false, false);
    }

    // C/D layout: VGPR r -> row (r + 8*half), lane l -> col
    float* crow = C + ((size_t)mtile * 16 + half * 8) * 64 + col0 + l;
#pragma unroll
    for (int r = 0; r < 8; ++r) crow[(size_t)r * 64] = c[r];
}

// ---------------- self-loop init: agg = dinv^2 * tmp ----------------

__global__ void k_selfloop(const float* __restrict__ dinv,
                           const float* __restrict__ tmp,
                           float* __restrict__ agg, int N) {
    int tid = blockIdx.x * blockDim.x + threadIdx.x;
    if (tid < N * HID) {
        int i = tid >> 6;
        float w = dinv[i];
        agg[tid] = w * w * tmp[tid];
    }
}

// ---------------- edge scatter into agg (layer 1) ----------------
// 16 consecutive lanes share one edge -> coalesced 64-float gather of tmp[src].

__global__ void k_scatter_agg(const long long* __restrict__ src,
                              const long long* __restrict__ dst,
                              const float* __restrict__ dinv,
                              const float* __restrict__ tmp,
                              float* __restrict__ agg, int E) {
    int tid = blockIdx.x * blockDim.x + threadIdx.x;
    int e = tid >> 4;
    int c = tid & 15;
    if (e >= E) return;
    int s = (int)src[e], d = (int)dst[e];
    float w = dinv[s] * dinv[d];
    float4 v = ((const float4*)(tmp + (size_t)s * HID))[c];
    float* o = agg + (size_t)d * HID + c * 4;
    atomicAdd(o + 0, w * v.x);
    atomicAdd(o + 1, w * v.y);
    atomicAdd(o + 2, w * v.z);
    atomicAdd(o + 3, w * v.w);
}

// ---------------- h = relu(agg + b1), in place ----------------

__global__ void k_bias_relu(float* __restrict__ agg,
                            const float* __restrict__ b1, int N) {
    int tid = blockIdx.x * blockDim.x + threadIdx.x;
    if (tid < N * HID) {
        float v = agg[tid] + b1[tid & 63];
        agg[tid] = v > 0.0f ? v : 0.0f;
    }
}

// ------- out init: bias + self-loop; tmp cols [0,32)=mu-pre, [32,64)=log-pre -------

__global__ void k_init_out(const float* __restrict__ dinv,
                           const float* __restrict__ tmp,
                           const float* __restrict__ bmu,
                           const float* __restrict__ blog,
                           float* __restrict__ out, int N) {
    int tid = blockIdx.x * blockDim.x + threadIdx.x;
    if (tid < N * HID) {
        int i = tid >> 6;
        int f = tid & 63;
        float w = dinv[i];
        float v = w * w * tmp[tid];
        if (f < 32) out[(size_t)i * 32 + f] = bmu[f] + v;
        else        out[(size_t)N * 32 + (size_t)i * 32 + (f - 32)] = blog[f - 32] + v;
    }
}

// ---------------- edge scatter into d_out (layer 2, mu + log) ----------------

__global__ void k_scatter_out(const long long* __restrict__ src,
                              const long long* __restrict__ dst,
                              const float* __restrict__ dinv,
                              const float* __restrict__ tmp,
                              float* __restrict__ out, int E, int N) {
    int tid = blockIdx.x * blockDim.x + threadIdx.x;
    int e = tid >> 4;
    int c = tid & 15;
    if (e >= E) return;
    int s = (int)src[e], d = (int)dst[e];
    float w = dinv[s] * dinv[d];
    float4 v = ((const float4*)(tmp + (size_t)s * HID))[c];
    int f = c * 4;  // 0..60; chunks 0-7 -> mu, 8-15 -> log (no straddling)
    float* o = (f < 32) ? (out + (size_t)d * 32 + f)
                        : (out + (size_t)N * 32 + (size_t)d * 32 + (f - 32));
    atomicAdd(o + 0, w * v.x);
    atomicAdd(o + 1, w * v.y);
    atomicAdd(o + 2, w * v.z);
    atomicAdd(o + 3, w * v.w);
}

// ---------------- launch ----------------

extern "C" void kernel_launch(void* const* d_in, const int* in_sizes, int n_in,
                              void* d_out, int out_size, void* d_ws, size_t ws_size,
                              hipStream_t stream) {
    const float*     x    = (const float*)d_in[0];
    const long long* ei   = (const long long*)d_in[1];   // int64 [2, E]
    const float*     W1   = (const float*)d_in[2];
    const float*     b1   = (const float*)d_in[3];
    const float*     Wmu  = (const float*)d_in[4];
    const float*     bmu  = (const float*)d_in[5];
    const float*     Wlog = (const float*)d_in[6];
    const float*     blog = (const float*)d_in[7];
    float*           out  = (float*)d_out;

    const int N = in_sizes[0] / IN_F;   // 100000 (divisible by 16)
    const int E = in_sizes[1] / 2;      // 1600000
    const long long* src = ei;
    const long long* dst = ei + E;

    // workspace: dinv[N] | tmp[N*64] | agg[N*64]  (~52 MB)
    float* dinv = (float*)d_ws;
    float* tmp  = dinv + 102400;              // padded past N, 256B aligned
    float* agg  = tmp + (size_t)N * HID;

    const int B = 256;
    const int nb_N   = (N + B - 1) / B;
    const int nb_E   = (E + B - 1) / B;
    const int nb_NH  = (N * HID + B - 1) / B;
    const int nb_E16 = (int)(((long long)E * 16 + B - 1) / B);

    // norm
    k_init_deg <<<nb_N, B, 0, stream>>>(dinv, N);
    k_count_deg<<<nb_E, B, 0, stream>>>(dst, dinv, E);
    k_rsqrt    <<<nb_N, B, 0, stream>>>(dinv, N);

    // layer 1: tmp = x @ W1 ; agg = scatter(norm * tmp[src]) incl. self-loops
    k_gemm_wmma<IN_F, false><<<N / 16, 128, 0, stream>>>(x, W1, nullptr, tmp);
    k_selfloop   <<<nb_NH,  B, 0, stream>>>(dinv, tmp, agg, N);
    k_scatter_agg<<<nb_E16, B, 0, stream>>>(src, dst, dinv, tmp, agg, E);
    k_bias_relu  <<<nb_NH,  B, 0, stream>>>(agg, b1, N);

    // layer 2: tmp = h @ [W_mu | W_log] ; scatter into d_out (bias+self-loop init)
    k_gemm_wmma<HID, true><<<N / 16, 128, 0, stream>>>(agg, Wmu, Wlog, tmp);
    k_init_out   <<<nb_NH,  B, 0, stream>>>(dinv, tmp, bmu, blog, out, N);
    k_scatter_out<<<nb_E16, B, 0, stream>>>(src, dst, dinv, tmp, out, E, N);
}